// InstancePinOptimizationArea_42700564857383
// MI455X (gfx1250) — compile-verified
//
#include <hip/hip_runtime.h>

// ---------------------------------------------------------------------------
// InstancePinOptimizationArea (DREAMPlace pin-utilization) for gfx1250.
// Phase 1: scatter-add separable 5x5 overlap contributions into a 512x512
//          f32 pin map (L2-resident, hardware global_atomic_add_f32).
// Phase 2: in-place clamp to utilization rates (float4 vectorized).
// Phase 3: gather 5x5 utilization window per movable node, bilinear reduce.
// NOTE: scatter kernel is defined FIRST so the disasm snippet shows the
// atomic lowering (expect global_atomic_add_f32, not a CAS loop).
// ---------------------------------------------------------------------------

namespace {
constexpr int   kNumNodes   = 1200000;
constexpr int   kNumPhys    = 1000000;   // NUM_NODES - NUM_FILLER
constexpr int   kNumMovable = 900000;
constexpr int   kNB         = 512;       // NBX == NBY
constexpr int   kNumBins    = kNB * kNB; // 262144
constexpr float kBS         = 1.953125f; // 1000/512, exact in fp32
constexpr float kStretch    = 1.4142135623730951f;
constexpr float kPinCapDen  = 1.953125f * 1.953125f * 0.05f; // BSX*BSY*UNIT_PIN_CAP
constexpr float kMaxRate    = 1.5f;
constexpr float kMinRate    = 1.0f / 1.5f;
constexpr int   K           = 5;
constexpr int   kBlock      = 256;       // 8 wave32 waves
}  // namespace

// Replicates reference _overlap_terms for one axis: 5 candidate bins starting
// at clamp(floor(lo/BS),0,NB-1); invalid (idx>=NB) columns get weight 0.
static __device__ __forceinline__ void overlap5(float lo, float hi,
                                                int idx[K], float ov[K]) {
  int il = (int)floorf(lo / kBS);
  il = min(max(il, 0), kNB - 1);
#pragma unroll
  for (int k = 0; k < K; ++k) {
    int id = il + k;
    const bool valid = id < kNB;
    id = valid ? id : (kNB - 1);
    const float blo = (float)id * kBS;
    float o = fminf(hi, blo + kBS) - fmaxf(lo, blo);
    o = fmaxf(o, 0.0f);
    idx[k] = id;
    ov[k] = valid ? o : 0.0f;
  }
}

__global__ __launch_bounds__(kBlock) void ipa_scatter_kernel(
    const float* __restrict__ pos, const float* __restrict__ nsx,
    const float* __restrict__ nsy, const int* __restrict__ flat,
    float* __restrict__ pin_map) {
  const int n = blockIdx.x * kBlock + threadIdx.x;
  if (n >= kNumPhys) return;

  // Prefetch the streamed inputs ahead (emits global_prefetch_b8).
  __builtin_prefetch(nsx + n + 4096, 0, 0);
  __builtin_prefetch(pos + n + 4096, 0, 0);

  const float sx = nsx[n];
  const float sy = nsy[n];
  const float hx = 0.5f * fmaxf(kBS * kStretch, sx);
  const float hy = 0.5f * fmaxf(kBS * kStretch, sy);
  const float cx = pos[n] + 0.5f * sx;
  const float cy = pos[kNumNodes + n] + 0.5f * sy;
  const float pw = (float)(flat[n + 1] - flat[n]);
  const float density = pw / (4.0f * hx * hy);

  int ix[K], iy[K];
  float ovx[K], ovy[K];
  overlap5(cx - hx, cx + hx, ix, ovx);
  overlap5(cy - hy, cy + hy, iy, ovy);

#pragma unroll
  for (int i = 0; i < K; ++i) {
    const float wxd = ovx[i] * density;
    if (wxd == 0.0f) continue;  // zero contribution: adding 0 is a no-op
    const int rowbase = ix[i] * kNB;
#pragma unroll
    for (int j = 0; j < K; ++j) {
      const float c = wxd * ovy[j];
      if (c != 0.0f) {
        // Hardware no-return f32 atomic (global_atomic_add_f32), L2 fast path.
        unsafeAtomicAdd(&pin_map[rowbase + iy[j]], c);
      }
    }
  }
}

__global__ __launch_bounds__(kBlock) void ipa_gather_kernel(
    const float* __restrict__ pos, const float* __restrict__ nsx,
    const float* __restrict__ nsy, const float* __restrict__ util,
    float* __restrict__ out) {
  const int n = blockIdx.x * kBlock + threadIdx.x;
  if (n >= kNumMovable) return;

  __builtin_prefetch(nsx + n + 4096, 0, 0);
  __builtin_prefetch(pos + n + 4096, 0, 0);

  const float xlo = pos[n];
  const float xhi = xlo + nsx[n];
  const float ylo = pos[kNumNodes + n];
  const float yhi = ylo + nsy[n];

  int ix[K], iy[K];
  float wx[K], wy[K];
  overlap5(xlo, xhi, ix, wx);
  overlap5(ylo, yhi, iy, wy);

  float acc = 0.0f;
#pragma unroll
  for (int i = 0; i < K; ++i) {
    if (wx[i] == 0.0f) continue;
    const int rowbase = ix[i] * kNB;
    float racc = 0.0f;
#pragma unroll
    for (int j = 0; j < K; ++j) {
      if (wy[j] != 0.0f) racc = fmaf(wy[j], util[rowbase + iy[j]], racc);
    }
    acc = fmaf(wx[i], racc, acc);
  }
  out[n] = acc;
}

// Zero the bin map, 4 bins per lane (kNumBins is a multiple of 4*kBlock).
__global__ __launch_bounds__(kBlock) void ipa_zero_kernel(
    float4* __restrict__ pin_map4) {
  const int b = blockIdx.x * kBlock + threadIdx.x;
  if (b < kNumBins / 4) pin_map4[b] = make_float4(0.f, 0.f, 0.f, 0.f);
}

// In-place pin_map -> util transform, 4 bins per lane.
__global__ __launch_bounds__(kBlock) void ipa_util_kernel(
    float4* __restrict__ pin_map4) {
  const int b = blockIdx.x * kBlock + threadIdx.x;
  if (b >= kNumBins / 4) return;
  float4 v = pin_map4[b];
  v.x = fminf(fmaxf(v.x / kPinCapDen, kMinRate), kMaxRate);
  v.y = fminf(fmaxf(v.y / kPinCapDen, kMinRate), kMaxRate);
  v.z = fminf(fmaxf(v.z / kPinCapDen, kMinRate), kMaxRate);
  v.w = fminf(fmaxf(v.w / kPinCapDen, kMinRate), kMaxRate);
  pin_map4[b] = v;
}

extern "C" void kernel_launch(void* const* d_in, const int* in_sizes, int n_in,
                              void* d_out, int out_size, void* d_ws,
                              size_t ws_size, hipStream_t stream) {
  (void)in_sizes; (void)n_in; (void)out_size; (void)ws_size;
  const float* pos  = (const float*)d_in[0];  // 2*NUM_NODES
  const float* nsx  = (const float*)d_in[1];  // NUM_NODES
  const float* nsy  = (const float*)d_in[2];  // NUM_NODES
  const int*   flat = (const int*)d_in[3];    // NUM_PHYS+1
  float* out     = (float*)d_out;             // NUM_MOVABLE
  float* pin_map = (float*)d_ws;              // kNumBins floats = 1 MiB

  const int gridBins4   = (kNumBins / 4 + kBlock - 1) / kBlock;
  const int gridScatter = (kNumPhys    + kBlock - 1) / kBlock;
  const int gridGather  = (kNumMovable + kBlock - 1) / kBlock;

  ipa_zero_kernel<<<gridBins4, kBlock, 0, stream>>>((float4*)pin_map);
  ipa_scatter_kernel<<<gridScatter, kBlock, 0, stream>>>(pos, nsx, nsy, flat,
                                                         pin_map);
  ipa_util_kernel<<<gridBins4, kBlock, 0, stream>>>((float4*)pin_map);
  ipa_gather_kernel<<<gridGather, kBlock, 0, stream>>>(pos, nsx, nsy, pin_map,
                                                       out);
}